// Attention_37606733644304
// MI455X (gfx1250) — compile-verified
//
#include <hip/hip_runtime.h>
#include <hip/hip_bf16.h>
#include <stdint.h>

typedef float v2f __attribute__((ext_vector_type(2)));
typedef float v4f __attribute__((ext_vector_type(4)));
typedef float v8f __attribute__((ext_vector_type(8)));

#define NB    16      // batch
#define NH    16      // heads
#define DHD   128     // head dim
#define KVLEN 4096    // cache length
#define NX    2048

// ---------------------------------------------------------------------------
// GEMM: out[16,N] = A[16,K] @ W[K,N] + bias[N], via V_WMMA_F32_16X16X4_F32.
// One wave per 16-wide N tile; 4 waves per block.
// A layout (16x4 f32): lanes 0-15 hold M=0..15 K={0,1}; lanes 16-31 K={2,3}.
// B layout (4x16 f32): lanes 0-15 hold N=0..15 K={0,1}; lanes 16-31 K={2,3}.
// C/D (16x16 f32, 8 VGPRs): lane%16 = N; VGPR r = M=r (+8 for lanes 16-31).
// ---------------------------------------------------------------------------
__global__ __launch_bounds__(128) void gemm16_f32(
    const float* __restrict__ A,    // [16, K] row-major
    const float* __restrict__ W,    // [K, N] row-major
    const float* __restrict__ bias, // [N]
    float* __restrict__ out,        // [16, N]
    int K, int N)
{
  const int wave = threadIdx.x >> 5;
  const int lane = threadIdx.x & 31;
  const int tile = blockIdx.x * 4 + wave;
  const int half = lane >> 4;      // K-pair selector
  const int idx  = lane & 15;      // M for A, N for B/C
  const int ncol = tile * 16 + idx;

  const float* arow = A + (size_t)idx * K + 2 * half;
  const float* wcol = W + (size_t)(2 * half) * N + ncol;

  v8f acc = {};
  #pragma unroll 8
  for (int k0 = 0; k0 < K; k0 += 4) {
    v2f a;
    a.x = arow[k0];
    a.y = arow[k0 + 1];
    v2f b;
    b.x = wcol[(size_t)k0 * N];
    b.y = wcol[(size_t)(k0 + 1) * N];
    acc = __builtin_amdgcn_wmma_f32_16x16x4_f32(
        /*neg_a=*/false, a, /*neg_b=*/false, b,
        /*c_mod=*/(short)0, acc, /*reuse_a=*/false, /*reuse_b=*/false);
  }

  const float bv = bias[ncol];
  #pragma unroll
  for (int r = 0; r < 8; ++r) {
    out[(size_t)(r + 8 * half) * N + ncol] = acc[r] + bv;
  }
}

// ---------------------------------------------------------------------------
// Overwrite the k-column / v-row at cache position `length` in the output
// copies of mem_k / mem_v. i enumerates b*2048 + h*128 + d.
// ---------------------------------------------------------------------------
__global__ __launch_bounds__(256) void cache_insert(
    const float* __restrict__ qkv,   // [16, 6144]
    const int* __restrict__ lenp,
    float* __restrict__ out_k,       // [B,H,DH,KV]
    float* __restrict__ out_v)       // [B,H,KV,DH]
{
  const int i = blockIdx.x * blockDim.x + threadIdx.x;
  if (i >= NB * NX) return;
  const int len = lenp[0];
  if (len < 0 || len >= KVLEN) return;

  const int b  = i >> 11;
  const int hd = i & (NX - 1);
  const int h  = hd >> 7;
  const int d  = hd & (DHD - 1);

  const float kval = qkv[b * (3 * NX) + NX + hd];
  const float vval = qkv[b * (3 * NX) + 2 * NX + hd];

  const size_t bh = (size_t)(b * NH + h);
  out_k[(bh * DHD + d) * KVLEN + len] = kval;
  out_v[(bh * KVLEN + len) * DHD + d] = vval;
}

// ---------------------------------------------------------------------------
// Single-token attention for one (b,h) per block. Scores -> masked softmax
// -> weighted V sum. Only positions [0, length) are unmasked (the newly
// inserted slot at index `length` is masked in the reference), so we read
// only the pre-existing cache.
// ---------------------------------------------------------------------------
__global__ __launch_bounds__(256) void attn_decode(
    const float* __restrict__ qkv,
    const float* __restrict__ mem_k,   // [B,H,DH,KV]
    const float* __restrict__ mem_v,   // [B,H,KV,DH]
    const int* __restrict__ lenp,
    float* __restrict__ a_heads)       // [B, H*DH]
{
  __shared__ float s_q[DHD];
  __shared__ float s_p[KVLEN];
  __shared__ float s_red[256];
  __shared__ float s_o[8 * DHD];

  const int bh = blockIdx.x;
  const int b  = bh >> 4;
  const int h  = bh & 15;
  const int t  = threadIdx.x;
  const int length = lenp[0];

  if (t < DHD) s_q[t] = qkv[b * (3 * NX) + h * DHD + t];
  __syncthreads();

  const float scale = 0.08838834764831845f;  // 1/sqrt(128)
  const float* kb = mem_k + (size_t)bh * DHD * KVLEN;

  // scores: each thread owns 4 consecutive n, stride 1024
  for (int n0 = t * 4; n0 < KVLEN; n0 += 1024) {
    if (n0 + 3 < length) {
      v4f acc = {0.f, 0.f, 0.f, 0.f};
      #pragma unroll 4
      for (int d = 0; d < DHD; ++d) {
        v4f kk = *reinterpret_cast<const v4f*>(kb + (size_t)d * KVLEN + n0);
        acc += s_q[d] * kk;
      }
      acc *= scale;
      *reinterpret_cast<v4f*>(&s_p[n0]) = acc;
    } else {
      for (int j = 0; j < 4; ++j) {
        const int n = n0 + j;
        float sc = -1e30f;
        if (n < length) {
          float a = 0.f;
          for (int d = 0; d < DHD; ++d) a += s_q[d] * kb[(size_t)d * KVLEN + n];
          sc = a * scale;
        }
        s_p[n] = sc;
      }
    }
  }
  __syncthreads();

  // max reduction
  float lmax = -1e30f;
  for (int n = t; n < KVLEN; n += 256) lmax = fmaxf(lmax, s_p[n]);
  s_red[t] = lmax;
  __syncthreads();
  for (int s = 128; s > 0; s >>= 1) {
    if (t < s) s_red[t] = fmaxf(s_red[t], s_red[t + s]);
    __syncthreads();
  }
  const float mx = s_red[0];
  __syncthreads();

  // exp + sum reduction
  float lsum = 0.f;
  for (int n = t; n < KVLEN; n += 256) {
    const float e = __expf(s_p[n] - mx);
    s_p[n] = e;
    lsum += e;
  }
  s_red[t] = lsum;
  __syncthreads();
  for (int s = 128; s > 0; s >>= 1) {
    if (t < s) s_red[t] += s_red[t + s];
    __syncthreads();
  }
  const float sum = s_red[0];
  const float inv = (sum > 0.f) ? (1.f / sum) : 0.f;

  // weighted V accumulation: 8 groups over n, 32 lanes x float4 over d
  const float* vb = mem_v + (size_t)bh * KVLEN * DHD;
  const int g  = t >> 5;
  const int d4 = (t & 31) * 4;
  v4f acc = {0.f, 0.f, 0.f, 0.f};
  for (int n = g; n < length; n += 8) {
    const float p = s_p[n];
    v4f vv = *reinterpret_cast<const v4f*>(vb + (size_t)n * DHD + d4);
    acc += p * vv;
  }
  #pragma unroll
  for (int j = 0; j < 4; ++j) s_o[g * DHD + d4 + j] = acc[j];
  __syncthreads();

  if (t < DHD) {
    float o = 0.f;
    #pragma unroll
    for (int gg = 0; gg < 8; ++gg) o += s_o[gg * DHD + t];
    a_heads[(size_t)bh * DHD + t] = o * inv;
  }
}

// ---------------------------------------------------------------------------
extern "C" void kernel_launch(void* const* d_in, const int* in_sizes, int n_in,
                              void* d_out, int out_size, void* d_ws, size_t ws_size,
                              hipStream_t stream) {
  (void)in_sizes; (void)n_in; (void)out_size; (void)ws_size;

  const float* x      = (const float*)d_in[0];  // [16,1,2048]
  const float* mem_k  = (const float*)d_in[1];  // [16,16,128,4096]
  const float* mem_v  = (const float*)d_in[2];  // [16,16,4096,128]
  const float* w_attn = (const float*)d_in[3];  // [2048,6144]
  const float* b_attn = (const float*)d_in[4];  // [6144]
  const float* w_proj = (const float*)d_in[5];  // [2048,2048]
  const float* b_proj = (const float*)d_in[6];  // [2048]
  const int*   lenp   = (const int*)d_in[7];    // scalar

  // output tuple layout: a [16,1,2048] | k_new [B,H,DH,KV] | v_new [B,H,KV,DH]
  float* out_a = (float*)d_out;
  float* out_k = out_a + (size_t)NB * NX;
  float* out_v = out_k + (size_t)NB * NH * DHD * KVLEN;

  // workspace: qkv [16,6144], a_heads [16,2048]
  float* qkv     = (float*)d_ws;
  float* a_heads = qkv + (size_t)NB * 3 * NX;

  // 1) bulk cache copies (the bandwidth floor: 2 x 537 MB each way)
  const size_t cache_bytes = (size_t)NB * NH * DHD * KVLEN * sizeof(float);
  hipMemcpyAsync(out_k, mem_k, cache_bytes, hipMemcpyDeviceToDevice, stream);
  hipMemcpyAsync(out_v, mem_v, cache_bytes, hipMemcpyDeviceToDevice, stream);

  // 2) fused QKV projection (WMMA f32): [16,2048] @ [2048,6144]
  gemm16_f32<<<(3 * NX / 16) / 4, 128, 0, stream>>>(x, w_attn, b_attn, qkv, NX, 3 * NX);

  // 3) insert new k column / v row at `length` into the output copies
  cache_insert<<<(NB * NX + 255) / 256, 256, 0, stream>>>(qkv, lenp, out_k, out_v);

  // 4) attention (memory-bound matvecs; one block per (b,h))
  attn_decode<<<NB * NH, 256, 0, stream>>>(qkv, mem_k, mem_v, lenp, a_heads);

  // 5) output projection (WMMA f32): [16,2048] @ [2048,2048]
  gemm16_f32<<<(NX / 16) / 4, 128, 0, stream>>>(a_heads, w_proj, b_proj, out_a, NX, NX);
}